// GCN_51187420233862
// MI455X (gfx1250) — compile-verified
//
#include <hip/hip_runtime.h>
#include <hip/hip_bf16.h>

typedef __attribute__((ext_vector_type(2))) float v2f;
typedef __attribute__((ext_vector_type(8))) float v8f;

#define FDIM 128     // feature width (F == H == 128 in reference)
#define CCLS 40      // class head cols
#define CDOM 2       // domain head cols
#define CPAD 48      // padded combined head cols (3 x 16)
#define ASTRIDE 132  // LDS A-panel row stride (pad 4 floats: conflict-free b64 reads)

// ---------------------------------------------------------------------------
// deg[i] = 1 (self loop)
__global__ void k_init_deg(float* __restrict__ deg, int n) {
    int i = blockIdx.x * blockDim.x + threadIdx.x;
    if (i < n) deg[i] = 1.0f;
}

// deg[dst[e]] += 1 per edge  (hardware global_atomic_add_f32)
__global__ void k_count(const int* __restrict__ dst, float* __restrict__ deg, int E) {
    int e = blockIdx.x * blockDim.x + threadIdx.x;
    if (e < E) unsafeAtomicAdd(&deg[dst[e]], 1.0f);
}

// deg -> rsqrt(deg)   (deg >= 1 guaranteed by self loops)
__global__ void k_rsqrt(float* __restrict__ deg, int n) {
    int i = blockIdx.x * blockDim.x + threadIdx.x;
    if (i < n) deg[i] = rsqrtf(deg[i]);
}

// Z[i][f] = X[i][f] * dinv[i]^2   (self-loop term; also zero-initializes Z)
__global__ void k_init_self(const float* __restrict__ X, const float* __restrict__ dinv,
                            float* __restrict__ Z, int n) {
    long long i = (long long)blockIdx.x * blockDim.x + threadIdx.x;
    long long total = (long long)n * FDIM;
    if (i >= total) return;
    int node = (int)(i >> 7);
    float di = dinv[node];
    Z[i] = X[i] * di * di;
}

// One wave per edge: Z[dst] += X[src] * dinv[src]*dinv[dst]
// 32 lanes x float4 = 128 floats per row; 4 hw f32 atomics per lane.
__global__ void k_scatter(const float* __restrict__ X, const int* __restrict__ src,
                          const int* __restrict__ dst, const float* __restrict__ dinv,
                          float* __restrict__ Z, int E) {
    int e = blockIdx.x * 8 + (threadIdx.x >> 5);
    if (e >= E) return;
    int lane = threadIdx.x & 31;
    int s = src[e];
    int d = dst[e];
    float norm = dinv[s] * dinv[d];
    const float4* xr = (const float4*)(X + (size_t)s * FDIM);
    float4 v = xr[lane];
    float* zr = Z + (size_t)d * FDIM + lane * 4;
    unsafeAtomicAdd(zr + 0, v.x * norm);
    unsafeAtomicAdd(zr + 1, v.y * norm);
    unsafeAtomicAdd(zr + 2, v.z * norm);
    unsafeAtomicAdd(zr + 3, v.w * norm);
}

// H = relu(Z @ W + b) : A [n x 128], W [128 x 128], WMMA f32 16x16x4.
// Block = 8 waves; A-panel (16x128) staged once in LDS; wave w -> col tile w.
__global__ void k_gemm_relu(const float* __restrict__ A, const float* __restrict__ W,
                            const float* __restrict__ bias, float* __restrict__ Hout, int n) {
    __shared__ float As[16 * ASTRIDE];
    int tid  = threadIdx.x;
    int row0 = blockIdx.x * 16;

    // cooperative panel load: 512 float4, 256 threads -> 2 each
#pragma unroll
    for (int t = 0; t < 2; ++t) {
        int idx = tid + t * 256;          // 0..511
        int i = idx >> 5;                 // panel row (32 float4 per row)
        int f = idx & 31;                 // float4 within row
        int r = row0 + i; if (r >= n) r = n - 1;
        *(float4*)(As + i * ASTRIDE + f * 4) =
            ((const float4*)A)[(size_t)r * (FDIM / 4) + f];
    }
    __syncthreads();

    int wave = tid >> 5;                  // 0..7 -> column tile
    int lane = tid & 31;
    int half = lane >> 4;
    int mn   = lane & 15;
    int col0 = wave * 16;

    v8f acc = {};
    const float* arow = As + mn * ASTRIDE + 2 * half;
    const float* wcol = W + (size_t)(2 * half) * FDIM + col0 + mn;

#pragma unroll
    for (int kb = 0; kb < FDIM; kb += 4) {
        v2f a, b;
        a.x = arow[kb];
        a.y = arow[kb + 1];
        b.x = wcol[(size_t)kb * FDIM];
        b.y = wcol[(size_t)(kb + 1) * FDIM];
        acc = __builtin_amdgcn_wmma_f32_16x16x4_f32(false, a, false, b,
                                                    (short)0, acc, false, false);
    }

    float bv = bias[col0 + mn];
#pragma unroll
    for (int j = 0; j < 8; ++j) {
        int rr = row0 + j + 8 * half;
        if (rr < n) {
            float v = acc[j] + bv;
            Hout[(size_t)rr * FDIM + col0 + mn] = v > 0.0f ? v : 0.0f;
        }
    }
}

// Pack W2 [128x40] and Wd [128x2] into Wc [128x48] (cols 42..47 zero) + bias.
__global__ void k_pack(const float* __restrict__ W2, const float* __restrict__ b2,
                       const float* __restrict__ Wd, const float* __restrict__ bd,
                       float* __restrict__ Wc, float* __restrict__ bc) {
    int i = blockIdx.x * blockDim.x + threadIdx.x;
    if (i < FDIM * CPAD) {
        int k = i / CPAD, c = i % CPAD;
        float v = 0.0f;
        if (c < CCLS)             v = W2[k * CCLS + c];
        else if (c < CCLS + CDOM) v = Wd[k * CDOM + (c - CCLS)];
        Wc[i] = v;
    } else if (i < FDIM * CPAD + CPAD) {
        int c = i - FDIM * CPAD;
        float v = 0.0f;
        if (c < CCLS)             v = b2[c];
        else if (c < CCLS + CDOM) v = bd[c - CCLS];
        bc[c] = v;
    }
}

// [y_class | y_domain] = Z @ Wc + bc, split-stored into d_out.
// Block = 3 waves (48 cols); one 16-row tile per block; LDS-staged A-panel.
__global__ void k_gemm_out(const float* __restrict__ A, const float* __restrict__ Wc,
                           const float* __restrict__ bc, float* __restrict__ out, int n) {
    __shared__ float As[16 * ASTRIDE];
    int tid  = threadIdx.x;
    int row0 = blockIdx.x * 16;

    for (int idx = tid; idx < 512; idx += 96) {
        int i = idx >> 5;
        int f = idx & 31;
        int r = row0 + i; if (r >= n) r = n - 1;
        *(float4*)(As + i * ASTRIDE + f * 4) =
            ((const float4*)A)[(size_t)r * (FDIM / 4) + f];
    }
    __syncthreads();

    int wave = tid >> 5;                  // 0..2 -> column tile
    int lane = tid & 31;
    int half = lane >> 4;
    int mn   = lane & 15;
    int col0 = wave * 16;

    v8f acc = {};
    const float* arow = As + mn * ASTRIDE + 2 * half;
    const float* wcol = Wc + (size_t)(2 * half) * CPAD + col0 + mn;

#pragma unroll
    for (int kb = 0; kb < FDIM; kb += 4) {
        v2f a, b;
        a.x = arow[kb];
        a.y = arow[kb + 1];
        b.x = wcol[(size_t)kb * CPAD];
        b.y = wcol[(size_t)(kb + 1) * CPAD];
        acc = __builtin_amdgcn_wmma_f32_16x16x4_f32(false, a, false, b,
                                                    (short)0, acc, false, false);
    }

    int c = col0 + mn;
    float bv = bc[c];
#pragma unroll
    for (int j = 0; j < 8; ++j) {
        int rr = row0 + j + 8 * half;
        if (rr < n) {
            float v = acc[j] + bv;
            if (c < CCLS)
                out[(size_t)rr * CCLS + c] = v;
            else if (c < CCLS + CDOM)
                out[(size_t)n * CCLS + (size_t)rr * CDOM + (c - CCLS)] = v;
        }
    }
}

// ---------------------------------------------------------------------------
extern "C" void kernel_launch(void* const* d_in, const int* in_sizes, int n_in,
                              void* d_out, int out_size, void* d_ws, size_t ws_size,
                              hipStream_t stream) {
    const float* x   = (const float*)d_in[0];
    const int*   ei  = (const int*)d_in[1];
    const float* W1  = (const float*)d_in[2];
    const float* b1  = (const float*)d_in[3];
    const float* W2  = (const float*)d_in[4];
    const float* b2  = (const float*)d_in[5];
    const float* Wd  = (const float*)d_in[6];
    const float* bd  = (const float*)d_in[7];
    float* out = (float*)d_out;

    const int N = in_sizes[0] / FDIM;
    const int E = in_sizes[1] / 2;
    const int* src = ei;
    const int* dst = ei + E;

    // workspace carve-up
    char* ws = (char*)d_ws;
    float* deg = (float*)ws;                          // N floats -> becomes dinv
    size_t off = (((size_t)N * 4) + 255) & ~(size_t)255;
    float* Z  = (float*)(ws + off); off += (size_t)N * FDIM * 4;   // agg buffer
    float* Hb = (float*)(ws + off); off += (size_t)N * FDIM * 4;   // hidden
    float* Wc = (float*)(ws + off); off += (size_t)FDIM * CPAD * 4;
    float* bc = (float*)(ws + off);
    (void)ws_size; (void)out_size; (void)n_in;

    const int T = 256;
    const long long nf = (long long)N * FDIM;

    // degree / normalization
    k_init_deg<<<(N + T - 1) / T, T, 0, stream>>>(deg, N);
    k_count<<<(E + T - 1) / T, T, 0, stream>>>(dst, deg, E);
    k_rsqrt<<<(N + T - 1) / T, T, 0, stream>>>(deg, N);   // deg now = dinv

    // pack output heads (independent)
    k_pack<<<(FDIM * CPAD + CPAD + T - 1) / T, T, 0, stream>>>(W2, b2, Wd, bd, Wc, bc);

    // layer 1: Z = Agg(x); H = relu(Z @ W1 + b1)
    k_init_self<<<(int)((nf + T - 1) / T), T, 0, stream>>>(x, deg, Z, N);
    k_scatter<<<(E + 7) / 8, T, 0, stream>>>(x, src, dst, deg, Z, E);
    k_gemm_relu<<<(N + 15) / 16, 256, 0, stream>>>(Z, W1, b1, Hb, N);

    // layer 2: Z = Agg(H); out = Z @ [W2|Wd] + [b2|bd]
    k_init_self<<<(int)((nf + T - 1) / T), T, 0, stream>>>(Hb, deg, Z, N);
    k_scatter<<<(E + 7) / 8, T, 0, stream>>>(Hb, src, dst, deg, Z, E);
    k_gemm_out<<<(N + 15) / 16, 96, 0, stream>>>(Z, Wc, bc, out, N);
}